// SMPLModel_87754771792345
// MI455X (gfx1250) — compile-verified
//
#include <hip/hip_runtime.h>
#include <math.h>

#define KJ 24
#define NVERT 6890
#define PREG 200
#define PPAD 208          // 13 * 16
#define KDIM 392          // 391 coeffs + 1 zero pad (multiple of 4)
#define NOUT 20670        // 3 * NVERT
#define MTILES 13
#define NGROUPS 323       // ceil(20670 / 64); each wave does 4 N-tiles of 16

typedef float v2f __attribute__((ext_vector_type(2)));
typedef float v8f __attribute__((ext_vector_type(8)));

__device__ const int d_PARENT[KJ] = {0,0,0,0,1,2,3,4,5,6,7,8,9,9,9,12,13,14,16,17,18,19,20,21};
__device__ const int d_LEN[KJ]    = {4,3,3,3,3,3,3,3,3,2,2,2,4,3,3,2,3,3,3,3,3,3,2,2};
__device__ const int d_NBR[KJ][4] = {
  {0,1,2,3},{0,1,4,0},{0,2,5,0},{0,3,6,0},{1,4,7,0},{2,5,8,0},{3,6,9,0},{4,7,10,0},
  {5,8,11,0},{6,9,0,0},{7,10,0,0},{8,11,0,0},{12,13,14,15},{12,13,16,0},{12,14,17,0},
  {12,15,0,0},{13,16,18,0},{14,17,19,0},{16,18,20,0},{17,19,21,0},{18,20,22,0},
  {19,21,23,0},{20,22,0,0},{21,23,0,0}};

__device__ __forceinline__ float blockReduce256(float v, float* s) {
  int t = threadIdx.x;
  s[t] = v; __syncthreads();
  for (int off = 128; off > 0; off >>= 1) {
    if (t < off) s[t] += s[t + off];
    __syncthreads();
  }
  float r = s[0];
  __syncthreads();
  return r;
}

// ---------------------------------------------------------------- init
__global__ void init_acc_kernel(float* acc) {
  if (threadIdx.x < 32) acc[threadIdx.x] = 0.f;
}

// ---------------------------------------------------------------- per-pose: quats->coeff, FK->G12
__global__ void pose_kernel(const float* __restrict__ J,      // (P,24,3)
                            const float* __restrict__ theta,  // (P,72)
                            const float* __restrict__ beta2,  // (1,)
                            float* __restrict__ G12,          // (P,24,12)
                            float* __restrict__ coeff)        // (PPAD,KDIM)
{
  int p = blockIdx.x * blockDim.x + threadIdx.x;
  if (p >= PPAD) return;
  if (p >= PREG) {                       // zero pad rows for the GEMM
    for (int i = 0; i < KDIM; ++i) coeff[p * KDIM + i] = 0.f;
    return;
  }
  const float b2 = beta2[0];
  const float* th = theta + p * 72;
  const float* Jp = J + p * KJ * 3;

  // ---- quaternions (axis-angle, w = cos(a/2)-1)
  float q[KJ][4];
  for (int k = 0; k < KJ; ++k) {
    float x = th[3*k], y = th[3*k+1], z = th[3*k+2];
    float n2 = x*x + y*y + z*z;
    float ang = sqrtf(fmaxf(n2, 1e-16f));
    float inv = 1.f / ang;
    float s = sinf(0.5f * ang);
    q[k][0] = x * inv * s; q[k][1] = y * inv * s; q[k][2] = z * inv * s;
    q[k][3] = cosf(0.5f * ang) - 1.f;
  }
  // ---- 391-entry coefficient row (qm flattened)
  for (int j = 0; j < 23; ++j) {
    int jj = j + 1;
    float v[17];
    for (int l = 0; l < 17; ++l) v[l] = 0.f;
    int len = d_LEN[jj];
    for (int i = 0; i < 4; ++i) {
      if (i < len) {
        int nb = d_NBR[jj][i];
        v[4*i+0] = q[nb][0]; v[4*i+1] = q[nb][1];
        v[4*i+2] = q[nb][2]; v[4*i+3] = q[nb][3];
      }
    }
    v[4 * len] += b2;                       // BETA_POS * beta_2
    int keep = 4 * d_LEN[j] + 1;            // SLICE_MASK row j
    for (int l = 0; l < 17; ++l)
      coeff[p * KDIM + j * 17 + l] = (l < keep) ? v[l] : 0.f;
  }
  coeff[p * KDIM + 391] = 0.f;              // K pad column (makes clamped B rows inert)

  // ---- forward kinematics (affine 3x4 chain)
  float G[KJ][12];
  for (int k = 0; k < KJ; ++k) {
    float x = th[3*k], y = th[3*k+1], z = th[3*k+2];
    float n2 = x*x + y*y + z*z;
    float ang = fmaxf(sqrtf(n2), 1e-8f);
    float ax = x / ang, ay = y / ang, az = z / ang;
    float c = cosf(ang), s = sinf(ang), oc = 1.f - c;
    float R[9];
    R[0]=c+oc*ax*ax;    R[1]=oc*ax*ay-s*az; R[2]=oc*ax*az+s*ay;
    R[3]=oc*ax*ay+s*az; R[4]=c+oc*ay*ay;    R[5]=oc*ay*az-s*ax;
    R[6]=oc*ax*az-s*ay; R[7]=oc*ay*az+s*ax; R[8]=c+oc*az*az;
    int par = d_PARENT[k];
    float tx = Jp[3*k]   - (k ? Jp[3*par]   : 0.f);
    float ty = Jp[3*k+1] - (k ? Jp[3*par+1] : 0.f);
    float tz = Jp[3*k+2] - (k ? Jp[3*par+2] : 0.f);
    if (k == 0) {
      for (int r = 0; r < 3; ++r) {
        G[0][r*4+0] = R[r*3+0]; G[0][r*4+1] = R[r*3+1]; G[0][r*4+2] = R[r*3+2];
      }
      G[0][3] = tx; G[0][7] = ty; G[0][11] = tz;
    } else {
      const float* P4 = G[par];
      for (int r = 0; r < 3; ++r) {
        float p0 = P4[r*4+0], p1 = P4[r*4+1], p2 = P4[r*4+2];
        G[k][r*4+0] = p0*R[0] + p1*R[3] + p2*R[6];
        G[k][r*4+1] = p0*R[1] + p1*R[4] + p2*R[7];
        G[k][r*4+2] = p0*R[2] + p1*R[5] + p2*R[8];
        G[k][r*4+3] = p0*tx + p1*ty + p2*tz + P4[r*4+3];
      }
    }
  }
  // subtract R_global * J and store
  for (int k = 0; k < KJ; ++k) {
    float jx = Jp[3*k], jy = Jp[3*k+1], jz = Jp[3*k+2];
    for (int r = 0; r < 3; ++r)
      G[k][r*4+3] -= G[k][r*4+0]*jx + G[k][r*4+1]*jy + G[k][r*4+2]*jz;
    for (int i = 0; i < 12; ++i)
      G12[(p * KJ + k) * 12 + i] = G[k][i];
  }
}

// ---------------------------------------------------------------- W_change + E_Wi + E_W
__global__ void wchange_kernel(const float* __restrict__ Wp, const float* __restrict__ Wi,
                               float* __restrict__ Wc, float* acc) {
  __shared__ float sred[256];
  int n = blockIdx.x * 256 + threadIdx.x;
  float ewi = 0.f, ew = 0.f;
  if (n < NVERT) {
    float w[KJ]; float ssum = 0.f;
    for (int k = 0; k < KJ; ++k) { float x = fmaxf(Wp[n*KJ+k], 0.f); w[k] = x; ssum += x; }
    float inv = 1.f / (ssum + 1e-8f);
    for (int k = 0; k < KJ; ++k) {
      float wc = w[k] * inv;
      Wc[n*KJ+k] = wc;
      float d = wc - Wi[n*KJ+k];
      ewi += d * d; ew += fabsf(wc);
    }
  }
  float r1 = blockReduce256(ewi, sred);
  float r2 = blockReduce256(ew, sred);
  if (threadIdx.x == 0) { atomicAdd(&acc[1], r1); atomicAdd(&acc[2], r2); }
}

// ---------------------------------------------------------------- E_A = sum |A|
__global__ void absA_kernel(const float* __restrict__ A, float* acc) {
  __shared__ float sred[256];
  const int total = 23 * NVERT;
  float s = 0.f;
  for (int i = blockIdx.x * 256 + threadIdx.x; i < total; i += gridDim.x * 256)
    s += fabsf(A[i]);
  float r = blockReduce256(s, sred);
  if (threadIdx.x == 0) atomicAdd(&acc[3], r);
}

// ---------------------------------------------------------------- per-slice sum K^2
__global__ void knorm_kernel(const float* __restrict__ Kf, float* acc) {
  __shared__ float sred[256];
  int j = blockIdx.x;                 // 0..22
  const int per = 17 * NOUT;
  const float* base = Kf + (size_t)j * per;
  float s = 0.f;
  for (int i = blockIdx.y * 256 + threadIdx.x; i < per; i += gridDim.y * 256) {
    float v = base[i]; s += v * v;
  }
  float r = blockReduce256(s, sred);
  if (threadIdx.x == 0) atomicAdd(&acc[4 + j], r);
}

// ---------------------------------------------------------------- WMMA GEMM:
// T_p[p,m] = sum_l coeff[p,l] * (K[l,m] * relu(A)[l/17, m/3]) + T[p,m]
// Branchless: K-pad handled via clamped row index (A column 391 is zero so the
// clamped B row contributes nothing); out-of-range output columns get garbage
// that stays confined to their own column of C and is dropped by the store guard.
__global__ __launch_bounds__(128)
void gemm_kernel(const float* __restrict__ coeff, // (PPAD,KDIM)
                 const float* __restrict__ Kf,    // (391,NOUT)
                 const float* __restrict__ A,     // (23,NVERT)
                 const float* __restrict__ T,     // (PREG,NOUT)
                 float* __restrict__ Tp)          // (PREG,NOUT)
{
  int wave = threadIdx.x >> 5;
  int lane = threadIdx.x & 31;
  int tile = blockIdx.x * 4 + wave;            // wave-uniform
  if (tile >= MTILES * NGROUPS) return;        // uniform exit, EXEC stays full
  int mt = tile % MTILES;                      // mt cycles fastest: block shares B cols
  int ng = tile / MTILES;                      // group of 4 consecutive N-tiles
  int half = lane >> 4;                        // 0: K rows {k,k+1}; 1: {k+2,k+3}
  int l16  = lane & 15;
  int arow = mt * 16 + l16;                    // pose row (padded rows are zero)

  int col[4], col3[4];
  for (int t = 0; t < 4; ++t) {
    int c = (ng * 4 + t) * 16 + l16;
    int cc = c < NOUT ? c : NOUT - 1;          // clamp for loads only
    col[t] = cc; col3[t] = cc / 3;
  }

  v8f Cacc[4];
  for (int t = 0; t < 4; ++t) Cacc[t] = (v8f){};

  for (int k4 = 0; k4 < KDIM; k4 += 4) {
    int ka = k4 + 2 * half;
    v2f afrag = *(const v2f*)&coeff[arow * KDIM + ka];   // 8B aligned (ka even)
    int l0 = ka;                                          // <= 390 always
    int l1 = ka + 1 <= 390 ? ka + 1 : 390;                // clamp pad row
    int j0 = l0 / 17, j1 = l1 / 17;
    const float* Krow0 = Kf + (size_t)l0 * NOUT;
    const float* Krow1 = Kf + (size_t)l1 * NOUT;
    const float* Arow0 = A + j0 * NVERT;
    const float* Arow1 = A + j1 * NVERT;
#pragma unroll
    for (int t = 0; t < 4; ++t) {
      v2f bfrag;
      bfrag.x = Krow0[col[t]] * fmaxf(Arow0[col3[t]], 0.f);
      bfrag.y = Krow1[col[t]] * fmaxf(Arow1[col3[t]], 0.f);
      Cacc[t] = __builtin_amdgcn_wmma_f32_16x16x4_f32(
          /*neg_a=*/false, afrag, /*neg_b=*/false, bfrag,
          /*c_mod=*/(short)0, Cacc[t], /*reuse_a=*/false, /*reuse_b=*/false);
    }
  }
  // C/D layout: VGPR v, lanes 0-15 -> M=v, N=lane; lanes 16-31 -> M=v+8, N=lane-16
#pragma unroll
  for (int t = 0; t < 4; ++t) {
    int c = (ng * 4 + t) * 16 + l16;
    if (c < NOUT) {
#pragma unroll
      for (int v = 0; v < 8; ++v) {
        int p = mt * 16 + v + 8 * half;
        if (p < PREG)
          Tp[(size_t)p * NOUT + c] = Cacc[t][v] + T[(size_t)p * NOUT + c];
      }
    }
  }
}

// ---------------------------------------------------------------- blend skinning + E_D
__global__ void skin_kernel(const float* __restrict__ G12, const float* __restrict__ Wc,
                            const float* __restrict__ Tp, const float* __restrict__ V,
                            float* __restrict__ verts, float* acc) {
  __shared__ float sG[KJ * 12];
  __shared__ float sred[256];
  int p = blockIdx.y;
  for (int i = threadIdx.x; i < KJ * 12; i += 256) sG[i] = G12[p * KJ * 12 + i];
  __syncthreads();
  int n = blockIdx.x * 256 + threadIdx.x;
  float ed = 0.f;
  if (n < NVERT) {
    float B[12];
    for (int i = 0; i < 12; ++i) B[i] = 0.f;
    for (int k = 0; k < KJ; ++k) {
      float w = Wc[n * KJ + k];
      const float* g = &sG[k * 12];
      for (int i = 0; i < 12; ++i) B[i] += w * g[i];
    }
    size_t base = (size_t)p * NOUT + (size_t)n * 3;
    float t0 = Tp[base + 0], t1 = Tp[base + 1], t2 = Tp[base + 2];
    for (int c = 0; c < 3; ++c) {
      float vert = B[c*4+0]*t0 + B[c*4+1]*t1 + B[c*4+2]*t2 + B[c*4+3];
      verts[base + c] = vert;
      float d = V[base + c] - vert;
      ed += d * d;
    }
  }
  float r = blockReduce256(ed, sred);
  if (threadIdx.x == 0) atomicAdd(&acc[0], r);
}

// ---------------------------------------------------------------- finalize E
__global__ void finalize_kernel(const float* acc, const int* epoch, float* out) {
  if (threadIdx.x == 0 && blockIdx.x == 0) {
    float e = (float)epoch[0];
    float g_wi = 0.1f   * expf(-0.1f   * e);
    float g_w  = 0.002f * expf(-0.008f * e);
    float g_a  = 0.001f * expf(-0.008f * e);
    float g_k  = 0.1f   * expf(-0.008f * e);
    float ek = 0.f;
    for (int j = 0; j < 23; ++j) ek += sqrtf(acc[4 + j]);
    out[0] = acc[0] + g_wi * acc[1] + g_w * acc[2] + g_a * acc[3] + g_k * ek;
  }
}

extern "C" void kernel_launch(void* const* d_in, const int* in_sizes, int n_in,
                              void* d_out, int out_size, void* d_ws, size_t ws_size,
                              hipStream_t stream) {
  const float* V     = (const float*)d_in[0];
  const float* T     = (const float*)d_in[1];
  const float* J     = (const float*)d_in[2];
  const float* theta = (const float*)d_in[3];
  const float* Wp    = (const float*)d_in[4];
  const float* Wi    = (const float*)d_in[5];
  const float* A     = (const float*)d_in[6];
  const float* Kf    = (const float*)d_in[7];
  const float* beta2 = (const float*)d_in[8];
  const int*   epoch = (const int*)d_in[9];
  float* out = (float*)d_out;

  float* ws    = (float*)d_ws;
  float* acc   = ws;                        // 32 floats
  float* G12   = ws + 32;                   // 200*24*12 = 57600
  float* coeff = G12 + PREG * KJ * 12;      // 208*392  = 81536
  float* Wc    = coeff + PPAD * KDIM;       // 6890*24  = 165360
  float* Tp    = Wc + NVERT * KJ;           // 200*20670 = 4134000

  init_acc_kernel<<<1, 32, 0, stream>>>(acc);
  pose_kernel<<<1, 256, 0, stream>>>(J, theta, beta2, G12, coeff);
  wchange_kernel<<<(NVERT + 255) / 256, 256, 0, stream>>>(Wp, Wi, Wc, acc);
  absA_kernel<<<64, 256, 0, stream>>>(A, acc);
  knorm_kernel<<<dim3(23, 32), 256, 0, stream>>>(Kf, acc);
  int nwaves = MTILES * NGROUPS;            // 4199
  gemm_kernel<<<(nwaves + 3) / 4, 128, 0, stream>>>(coeff, Kf, A, T, Tp);
  skin_kernel<<<dim3((NVERT + 255) / 256, PREG), 256, 0, stream>>>(G12, Wc, Tp, V, out + 1, acc);
  finalize_kernel<<<1, 32, 0, stream>>>(acc, epoch, out);
}